// C3DNet_10814727652007
// MI455X (gfx1250) — compile-verified
//
#include <hip/hip_runtime.h>
#include <hip/hip_bf16.h>

typedef __attribute__((ext_vector_type(2))) float v2f;
typedef __attribute__((ext_vector_type(8))) float v8f;

// Fast sigmoid: v_exp_f32 + v_rcp_f32 (both TRANS-pipe ops) instead of the
// IEEE divide expansion (v_div_scale/v_div_fmas chains seen in round 1).
__device__ __forceinline__ float sigmoidf_(float v) {
    return __builtin_amdgcn_rcpf(1.0f + __expf(-v));
}

__device__ __forceinline__ v8f wmma_f32_k4(v2f a, v2f b, v8f c) {
    // V_WMMA_F32_16X16X4_F32 : D(16x16,f32) = A(16x4,f32) x B(4x16,f32) + C
    return __builtin_amdgcn_wmma_f32_16x16x4_f32(
        /*neg_a=*/false, a, /*neg_b=*/false, b,
        /*c_mod=*/(short)0, c, /*reuse_a=*/false, /*reuse_b=*/false);
}

// ---------------------------------------------------------------------------
// Kernel 1: per-sample conv1 -> sigmoid -> conv2 -> sigmoid -> avgpool(2)
// x: (10, 52, 7, 24) f32.  Output h3: (10, 27) f32 in workspace.
// One block per sample; all intermediates live in LDS.
// ---------------------------------------------------------------------------
__global__ void __launch_bounds__(256) conv_stage_kernel(
    const float* __restrict__ x,
    const float* __restrict__ w1, const float* __restrict__ b1,
    const float* __restrict__ w2, const float* __restrict__ b2,
    float* __restrict__ h3)
{
    __shared__ float s_h1[24 * 6 * 12];   // conv1 out (d,h,w)
    __shared__ float s_h2[6 * 6 * 6];     // conv2 out
    __shared__ float s_w1[24];
    __shared__ float s_w2[8];

    const int b   = blockIdx.x;
    const int tid = threadIdx.x;
    const float* __restrict__ xb = x + b * 8736;   // 52*7*24

    if (tid < 24)       s_w1[tid]      = w1[tid];
    else if (tid < 32)  s_w2[tid - 24] = w2[tid - 24];
    __syncthreads();

    // conv1: kernel (6,2,2), stride (2,1,2), out (24,6,12)
    const float bias1 = b1[0];
    for (int idx = tid; idx < 1728; idx += 256) {
        const int d = idx / 72, r = idx % 72, h = r / 12, w = r % 12;
        float s = bias1;
        #pragma unroll
        for (int kd = 0; kd < 6; ++kd)
            #pragma unroll
            for (int kh = 0; kh < 2; ++kh)
                #pragma unroll
                for (int kw = 0; kw < 2; ++kw)
                    s += xb[(2*d + kd)*168 + (h + kh)*24 + (2*w + kw)]
                         * s_w1[kd*4 + kh*2 + kw];
        s_h1[idx] = sigmoidf_(s);
    }
    __syncthreads();

    // conv2: kernel (4,1,2), stride (4,1,2), out (6,6,6)
    const float bias2 = b2[0];
    if (tid < 216) {
        const int d = tid / 36, r = tid % 36, h = r / 6, w = r % 6;
        float s = bias2;
        #pragma unroll
        for (int kd = 0; kd < 4; ++kd)
            #pragma unroll
            for (int kw = 0; kw < 2; ++kw)
                s += s_h1[(4*d + kd)*72 + h*12 + (2*w + kw)] * s_w2[kd*2 + kw];
        s_h2[tid] = sigmoidf_(s);
    }
    __syncthreads();

    // avgpool k=2 -> (3,3,3) -> 27 features
    if (tid < 27) {
        const int d = tid / 9, r = tid % 9, h = r / 3, w = r % 3;
        float s = 0.0f;
        #pragma unroll
        for (int i = 0; i < 2; ++i)
            #pragma unroll
            for (int j = 0; j < 2; ++j)
                #pragma unroll
                for (int l = 0; l < 2; ++l)
                    s += s_h2[(2*d + i)*36 + (2*h + j)*6 + (2*w + l)];
        h3[b*27 + tid] = s * 0.125f;
    }
}

// ---------------------------------------------------------------------------
// FC layer via V_WMMA_F32_16X16X4_F32.
// A (LDS): 16 x Kpad row-major (rows >= batch are zero/garbage -> only pollute
// D rows that are never stored).  W (global, row-major (N, Kreal)): B[k][n] =
// W[n*Kreal + k].  out = sigmoid(A @ W^T + bias), rows < mlimit stored.
// Per-wave: one 16x16 output tile, waves stride over N-tiles.
// ---------------------------------------------------------------------------
static __device__ void fc_layer(const float* __restrict__ A, int lda,
                                int Kpad, int Kreal,
                                const float* __restrict__ W,
                                const float* __restrict__ bias,
                                int N, float* __restrict__ out, int ldo,
                                int mlimit, int wave, int lane, int nwaves)
{
    const int m      = lane & 15;          // A row
    const int kboff  = (lane >> 4) << 1;   // K sub-offset within 4-step
    const int mbase  = (lane >> 4) << 3;   // D row base for this half-wave
    const int ntiles = (N + 15) >> 4;

    for (int t = wave; t < ntiles; t += nwaves) {
        const int n = (t << 4) + (lane & 15);
        v8f acc = {};
        for (int k0 = 0; k0 < Kpad; k0 += 4) {
            const int kb = k0 + kboff;
            v2f af;
            af.x = A[m*lda + kb];
            af.y = A[m*lda + kb + 1];
            v2f bf;
            bf.x = (n < N && kb     < Kreal) ? W[n*Kreal + kb    ] : 0.0f;
            bf.y = (n < N && kb + 1 < Kreal) ? W[n*Kreal + kb + 1] : 0.0f;
            acc = wmma_f32_k4(af, bf, acc);
        }
        if (n < N) {
            const float bv = bias[n];
            #pragma unroll
            for (int r = 0; r < 8; ++r) {
                const int mr = mbase + r;
                if (mr < mlimit) out[mr*ldo + n] = sigmoidf_(acc[r] + bv);
            }
        }
    }
}

// ---------------------------------------------------------------------------
// Kernel 2: fc4 (27->80) -> fc5 (80->200) -> fc6 (200->676) -> block-diagonal
// einsum: per sample b, out[b] = h6[b].reshape(13,52) @ x[b].reshape(52,168).
// Single workgroup (8 wave32s); all activations in LDS (~46 KB < 64 KB).
// ---------------------------------------------------------------------------
__global__ void __launch_bounds__(256) fc_einsum_kernel(
    const float* __restrict__ h3,
    const float* __restrict__ w4, const float* __restrict__ b4,
    const float* __restrict__ w5, const float* __restrict__ b5,
    const float* __restrict__ w6, const float* __restrict__ b6,
    const float* __restrict__ x,
    float* __restrict__ out)
{
    __shared__ float sA [16 * 28];    // h3, K padded 27->28 with zeros
    __shared__ float sH4[16 * 80];
    __shared__ float sH5[16 * 200];
    __shared__ float sH6[10 * 676];

    const int tid    = threadIdx.x;
    const int lane   = tid & 31;      // wave32
    const int wave   = tid >> 5;
    const int nwaves = 256 >> 5;      // 8

    for (int i = tid; i < 16 * 28; i += 256) {
        const int bm = i / 28, k = i % 28;
        sA[i] = (bm < 10 && k < 27) ? h3[bm*27 + k] : 0.0f;
    }
    __syncthreads();

    fc_layer(sA,  28,  28,  27, w4, b4,  80, sH4,  80, 16, wave, lane, nwaves);
    __syncthreads();
    fc_layer(sH4, 80,  80,  80, w5, b5, 200, sH5, 200, 16, wave, lane, nwaves);
    __syncthreads();
    fc_layer(sH5, 200, 200, 200, w6, b6, 676, sH6, 676, 10, wave, lane, nwaves);
    __syncthreads();

    // einsum: per sample, D(13x168) = A(13x52) @ B(52x168); pad M 13->16,
    // tile N in 11 tiles of 16 (last tile 8 valid cols).
    const int m     = lane & 15;
    const int kboff = (lane >> 4) << 1;
    const int mbase = (lane >> 4) << 3;
    const int nt    = 11;
    for (int task = wave; task < 10 * nt; task += nwaves) {
        const int b = task / nt;
        const int n = (task % nt) * 16 + (lane & 15);
        const float* __restrict__ xb = x + b * 8736;    // weeks[b] (52,168)
        const float* __restrict__ hb = sH6 + b * 676;   // h6[b] as (13,52)
        v8f acc = {};
        for (int k0 = 0; k0 < 52; k0 += 4) {
            const int kb = k0 + kboff;
            v2f af;
            af.x = (m < 13) ? hb[m*52 + kb    ] : 0.0f;
            af.y = (m < 13) ? hb[m*52 + kb + 1] : 0.0f;
            v2f bf;
            bf.x = (n < 168) ? xb[kb*168 + n]       : 0.0f;
            bf.y = (n < 168) ? xb[(kb + 1)*168 + n] : 0.0f;
            acc = wmma_f32_k4(af, bf, acc);
        }
        if (n < 168) {
            #pragma unroll
            for (int r = 0; r < 8; ++r) {
                const int mr = mbase + r;
                if (mr < 13) out[b*2184 + mr*168 + n] = acc[r];
            }
        }
    }
}

// ---------------------------------------------------------------------------
extern "C" void kernel_launch(void* const* d_in, const int* in_sizes, int n_in,
                              void* d_out, int out_size, void* d_ws, size_t ws_size,
                              hipStream_t stream) {
    const float* x  = (const float*)d_in[0];
    const float* w1 = (const float*)d_in[1];
    const float* b1 = (const float*)d_in[2];
    const float* w2 = (const float*)d_in[3];
    const float* b2 = (const float*)d_in[4];
    const float* w4 = (const float*)d_in[5];
    const float* b4 = (const float*)d_in[6];
    const float* w5 = (const float*)d_in[7];
    const float* b5 = (const float*)d_in[8];
    const float* w6 = (const float*)d_in[9];
    const float* b6 = (const float*)d_in[10];
    float* out = (float*)d_out;
    float* h3  = (float*)d_ws;     // 10*27 floats of scratch

    conv_stage_kernel<<<10, 256, 0, stream>>>(x, w1, b1, w2, b2, h3);
    fc_einsum_kernel<<<1, 256, 0, stream>>>(h3, w4, b4, w5, b5, w6, b6, x, out);
}